// JointConRanFieldNegev_73701638800082
// MI455X (gfx1250) — compile-verified
//
#include <hip/hip_runtime.h>

typedef __attribute__((ext_vector_type(2))) float v2f;
typedef __attribute__((ext_vector_type(8))) float v8f;

#define BN 2
#define HW 112                   // downsampled H=W
#define NPIX (HW * HW)           // 12544 pixels per batch item
#define NTILE (NPIX / 16)        // 784 16-row tiles
#define HTILE (NTILE / 2)        // 392 row-pairs (balanced triangle)
#define LOG2E 1.4426950408889634f
#define NEG_HALF_LOG2E (-0.72134752044448170f)  // -0.5 * log2(e)

// ---------------------------------------------------------------------------
// Kernel 1: per-pixel prep. Packs each pixel as an 8-float record laid out so
// a lane's whole WMMA fragment half (F and S, K-pair k0=2*hi) is ONE b128:
//   rec[0..3] = { F0, F1, S0, S1 }   (lanes 0-15,  K=0,1)
//   rec[4..7] = { F2, 0,  0,  0  }   (lanes 16-31, K=2,3; zero-padded)
// SQ2 = -0.5*log2(e)*||f||^2 pre-scaled so the main loop needs no pk_mul.
// Also zeroes the scalar output (harness poisons d_out).
// ---------------------------------------------------------------------------
__global__ void crf_prep_kernel(const float* __restrict__ fcams,
                                const float* __restrict__ raw,
                                float* __restrict__ Pbuf,   // [BN*NPIX] 8 floats
                                float* __restrict__ SQ2,    // [BN*NPIX]
                                float* __restrict__ out) {
  int tid = blockIdx.x * blockDim.x + threadIdx.x;
  if (tid == 0) out[0] = 0.0f;
  if (tid >= BN * NPIX) return;

  int b = tid / NPIX;
  int p = tid - b * NPIX;
  int y = p / HW;
  int x = p - y * HW;
  int Y = 2 * y, X = 2 * x;

  // nearest-downsampled image, features = rgb / sigma_rgb
  const float inv_sigma = 1.0f / 15.0f;
  float f0 = raw[((b * 3 + 0) * 224 + Y) * 224 + X] * inv_sigma;
  float f1 = raw[((b * 3 + 1) * 224 + Y) * 224 + X] * inv_sigma;
  float f2 = raw[((b * 3 + 2) * 224 + Y) * 224 + X] * inv_sigma;
  SQ2[tid] = NEG_HALF_LOG2E * (f0 * f0 + f1 * f1 + f2 * f2);

  // 2-channel softmax (p = 1/(1+exp(x1-x0))) + 2x2-mean bilinear downsample
  const float* c0 = fcams + (size_t)((b * 2 + 0) * 224) * 224;
  const float* c1 = fcams + (size_t)((b * 2 + 1) * 224) * 224;
  float pbar = 0.0f;
#pragma unroll
  for (int dy = 0; dy < 2; ++dy) {
#pragma unroll
    for (int dx = 0; dx < 2; ++dx) {
      int off = (Y + dy) * 224 + (X + dx);
      float d = c1[off] - c0[off];
      float t = __builtin_amdgcn_exp2f(d * LOG2E);
      pbar += __builtin_amdgcn_rcpf(1.0f + t);
    }
  }
  pbar *= 0.25f;

  float4* rec = reinterpret_cast<float4*>(Pbuf + (size_t)tid * 8);
  rec[0] = make_float4(f0, f1, pbar, 1.0f - pbar);  // low-half fragment (K=0,1)
  rec[1] = make_float4(f2, 0.0f, 0.0f, 0.0f);       // high-half fragment (K=2,3)
}

// ---------------------------------------------------------------------------
// Kernel 2: fused symmetric CRF energy. tile(jt,it) == tile(it,jt), so only
// jt >= it is computed; off-diagonal tiles are doubled for free by adding
// +1.0 to the exp2 argument (2*exp2(a) == exp2(a+1)). Each block processes
// the balanced row-pair {it, NTILE-1-it} -> constant NTILE+1 j-tiles/block.
// Per j-tile (one wave, EXEC all-ones):
//   G = F_i x F_j^T ,  W = S_i x S_j^T     (2x v_wmma_f32_16x16x4_f32)
//   acc += exp2(log2e*G + sqi2 + sqj2 + bias) * W   (8 elem/lane, v_exp_f32)
// C/D layout: VGPR r, lane c -> M = r + 8*(c>=16), N = c&15.
// ---------------------------------------------------------------------------
__global__ void crf_energy_kernel(const float* __restrict__ Pbuf,
                                  const float* __restrict__ SQ2,
                                  float* __restrict__ out) {
  int blk = blockIdx.x;
  int b = blk / HTILE;
  int itp = blk - b * HTILE;
  int lane = threadIdx.x & 31;
  int wave = threadIdx.x >> 5;
  int hi = lane >> 4;   // fragment half: K-pair 2*hi
  int m = lane & 15;    // row (A) / col (B,C,D) index within tile
  int base = b * NPIX;

  v8f acc = {};
  const v8f zeroC = {};
  int rows[2] = {itp, NTILE - 1 - itp};

#pragma unroll
  for (int ph = 0; ph < 2; ++ph) {
    int it = rows[ph];
    int i0 = it * 16;

    // A fragments (F and S halves in one b128)
    float4 ra = *(reinterpret_cast<const float4*>(
                      Pbuf + (size_t)(base + i0 + m) * 8) + hi);
    v2f aF = {ra.x, ra.y};
    v2f aS = {ra.z, ra.w};
    float sqi2[8];
#pragma unroll
    for (int r = 0; r < 8; ++r) sqi2[r] = SQ2[base + i0 + r + 8 * hi];

    for (int jt = it + wave; jt < NTILE; jt += 8) {
      int j0 = jt * 16;
      float4 rb = *(reinterpret_cast<const float4*>(
                        Pbuf + (size_t)(base + j0 + m) * 8) + hi);
      v2f bF = {rb.x, rb.y};
      v2f bS = {rb.z, rb.w};
      // +1.0 doubles off-diagonal tiles (symmetry); sq pre-scaled by -ln2e/2
      float sqjb = SQ2[base + j0 + m] + ((jt == it) ? 0.0f : 1.0f);

      // (neg_a, A, neg_b, B, c_mod, C, reuse_a, reuse_b)
      v8f G = __builtin_amdgcn_wmma_f32_16x16x4_f32(
          false, aF, false, bF, (short)0, zeroC, false, false);
      v8f W = __builtin_amdgcn_wmma_f32_16x16x4_f32(
          false, aS, false, bS, (short)0, zeroC, false, false);

#pragma unroll
      for (int r = 0; r < 8; ++r) {
        float arg = __builtin_fmaf(LOG2E, G[r], sqi2[r] + sqjb);
        float k = __builtin_amdgcn_exp2f(arg);     // exp(-0.5||fi-fj||^2)*wt
        acc[r] = __builtin_fmaf(k, W[r], acc[r]);
      }
    }
  }

  float local = 0.0f;
#pragma unroll
  for (int r = 0; r < 8; ++r) local += acc[r];
  // wave32 reduction
#pragma unroll
  for (int off = 16; off > 0; off >>= 1) local += __shfl_xor(local, off, 32);
  if (lane == 0) atomicAdd(out, -0.5f * local);  // WEIGHT * (-sum / B), B=2
}

// ---------------------------------------------------------------------------
extern "C" void kernel_launch(void* const* d_in, const int* in_sizes, int n_in,
                              void* d_out, int out_size, void* d_ws,
                              size_t ws_size, hipStream_t stream) {
  (void)in_sizes; (void)n_in; (void)out_size; (void)ws_size;
  const float* fcams = (const float*)d_in[0];  // [2,2,224,224] f32
  const float* raw   = (const float*)d_in[1];  // [2,3,224,224] f32
  float* out = (float*)d_out;                  // scalar f32

  float* Pbuf = (float*)d_ws;                  // BN*NPIX * 8 floats (~803 KB)
  float* SQ2  = Pbuf + (size_t)BN * NPIX * 8;  // BN*NPIX floats    (~100 KB)

  crf_prep_kernel<<<(BN * NPIX + 255) / 256, 256, 0, stream>>>(
      fcams, raw, Pbuf, SQ2, out);
  crf_energy_kernel<<<BN * HTILE, 256, 0, stream>>>(Pbuf, SQ2, out);
}